// SparseSubpixelExpert_9818295239053
// MI455X (gfx1250) — compile-verified
//
#include <hip/hip_runtime.h>
#include <hip/hip_bf16.h>
#include <math.h>

#define BATCH 8
#define C1 256
#define IMH 192
#define IMW 192
#define PP 4
#define GH 48
#define GW 48
#define NPATCH (GH*GW)          // 2304
#define KSEL 576                // ceil(2304*0.25)
#define C4 1024
#define HID 128
#define RH 64
#define BN_EPS 1e-5f

typedef __attribute__((ext_vector_type(2))) float v2f;
typedef __attribute__((ext_vector_type(8))) float v8f;

__device__ __forceinline__ float silu_f(float v) {
    return v / (1.0f + __expf(-v));
}

// ---------------------------------------------------------------------------
// Kernel 1: 4x4 average pooling, write pooled transposed: pooled[patch][C1]
// ---------------------------------------------------------------------------
__global__ __launch_bounds__(256) void pool_kernel(const float* __restrict__ x,
                                                   float* __restrict__ pooled) {
    int idx = blockIdx.x * 256 + threadIdx.x;        // B*C1*GH*GW threads
    int gx = idx % GW; int t = idx / GW;
    int gy = t % GH;   t /= GH;
    int c  = t % C1;   int b = t / C1;
    const float* src = x + (((size_t)(b * C1 + c)) * IMH + gy * PP) * IMW + gx * PP;
    float s = 0.0f;
    #pragma unroll
    for (int r = 0; r < 4; r++) {
        float4 v = *(const float4*)(src + r * IMW);
        s += v.x + v.y + v.z + v.w;
    }
    pooled[((size_t)(b * GH + gy) * GW + gx) * C1 + c] = s * (1.0f / 16.0f);
}

// ---------------------------------------------------------------------------
// Kernel 2: router MLP, one wave32 per patch
// ---------------------------------------------------------------------------
__global__ __launch_bounds__(256) void router_kernel(const float* __restrict__ pooled,
                                                     const float* __restrict__ rw1,
                                                     const float* __restrict__ rg1,
                                                     const float* __restrict__ rb1,
                                                     const float* __restrict__ rw2,
                                                     const float* __restrict__ rbias2,
                                                     float* __restrict__ scores) {
    int wave = threadIdx.x >> 5;
    int lane = threadIdx.x & 31;
    int pid  = blockIdx.x * 8 + wave;                // < BATCH*NPATCH
    const float* pv = pooled + (size_t)pid * C1;
    const float* w0 = rw1 + (size_t)lane * C1;
    const float* w1 = rw1 + (size_t)(lane + 32) * C1;
    float a0 = 0.0f, a1 = 0.0f;
    for (int c = 0; c < C1; c++) {
        float p = pv[c];
        a0 = fmaf(w0[c], p, a0);
        a1 = fmaf(w1[c], p, a1);
    }
    const float inv = rsqrtf(1.0f + BN_EPS);
    float h0 = silu_f(a0 * (rg1[lane] * inv) + rb1[lane]);
    float h1 = silu_f(a1 * (rg1[lane + 32] * inv) + rb1[lane + 32]);
    float part = rw2[lane] * h0 + rw2[lane + 32] * h1;
    #pragma unroll
    for (int off = 16; off > 0; off >>= 1)
        part += __shfl_xor(part, off, 32);
    if (lane == 0)
        scores[pid] = 1.0f / (1.0f + __expf(-(part + rbias2[0])));
}

// ---------------------------------------------------------------------------
// Kernel 3: exact top-k by rank counting (matches jax.lax.top_k tie-breaking:
// among equal scores the lower index ranks first). One block per batch.
// ---------------------------------------------------------------------------
__global__ __launch_bounds__(256) void topk_kernel(const float* __restrict__ scores,
                                                   int* __restrict__ sel_idx,
                                                   float* __restrict__ sel_val) {
    __shared__ float s[NPATCH];
    __shared__ int cnt;
    int b = blockIdx.x;
    if (threadIdx.x == 0) cnt = 0;
    for (int i = threadIdx.x; i < NPATCH; i += 256)
        s[i] = scores[(size_t)b * NPATCH + i];
    __syncthreads();
    for (int i = threadIdx.x; i < NPATCH; i += 256) {
        float si = s[i];
        int rank = 0;
        for (int j = 0; j < NPATCH; j++) {
            float sj = s[j];
            rank += (sj > si) || (sj == si && j < i);
        }
        if (rank < KSEL) {
            int slot = atomicAdd(&cnt, 1);
            sel_idx[(size_t)b * KSEL + slot] = i;
            sel_val[(size_t)b * KSEL + slot] = si;
        }
    }
}

// ---------------------------------------------------------------------------
// Kernel 4: expert. One workgroup (8 waves / 256 thr) = 4 patches = N-tile 16.
// GEMM1: [128,1024]x[1024,16] via v_wmma_f32_16x16x4_f32 (one M-tile/wave).
// DW 3x3 on 2x2 grid = per-channel 4x4 linear map. GEMM2: [1024,128]x[128,16].
// Epilogue: BN, score scale, pixel-shuffle, scatter store.
// ---------------------------------------------------------------------------
__global__ __launch_bounds__(256) void expert_kernel(const float* __restrict__ x,
                                                     const int* __restrict__ sel_idx,
                                                     const float* __restrict__ sel_val,
                                                     const float* __restrict__ ew1,
                                                     const float* __restrict__ eg1,
                                                     const float* __restrict__ eb1,
                                                     const float* __restrict__ dww,
                                                     const float* __restrict__ dwg,
                                                     const float* __restrict__ dwb,
                                                     const float* __restrict__ ew2,
                                                     const float* __restrict__ eg2,
                                                     const float* __restrict__ eb2,
                                                     float* __restrict__ out) {
    __shared__ float A1[C4 * 16];    // 64 KB: pixel-unshuffled activations [K][N]
    __shared__ float Hs[HID * 16];   // GEMM1 output after BN+SiLU
    __shared__ float A2[HID * 16];   // DW output after BN+SiLU (GEMM2 B matrix)
    __shared__ float wdw[HID * 9];
    __shared__ int   pidx[4];
    __shared__ float pval[4];
    __shared__ int   py0s[4], px0s[4];

    const int t = threadIdx.x;
    const int b = (blockIdx.x * 4) / KSEL;           // KSEL % 4 == 0

    if (t < 4) {
        int e  = blockIdx.x * 4 + t;
        int pi = sel_idx[e];
        pidx[t] = pi;
        pval[t] = sel_val[e];
        py0s[t] = (pi / GW) * PP;
        px0s[t] = (pi % GW) * PP;
    }
    for (int i = t; i < HID * 9; i += 256) wdw[i] = dww[i];
    __syncthreads();

    // ---- gather + pixel_unshuffle into A1[k][n] -------------------------
    // k = c*4 + ry*2 + rx ; n = p*4 + oh*2 + ow ; pixel (oh*2+ry, ow*2+rx)
    #pragma unroll
    for (int r = 0; r < 16; r++) {
        int row_id = t + r * 256;                    // 0..4095 = (c, p, y)
        int c   = row_id >> 4;
        int rem = row_id & 15;
        int p   = rem >> 2;
        int y   = rem & 3;
        const float* src = x + (((size_t)(b * C1 + c)) * IMH + (py0s[p] + y)) * IMW + px0s[p];
        float4 v = *(const float4*)src;
        int oh = y >> 1, ry = y & 1;
        int kb = c * 4 + ry * 2;
        int nb = p * 4 + oh * 2;
        A1[kb * 16 + nb]           = v.x;   // x=0: ow=0,rx=0
        A1[(kb + 1) * 16 + nb]     = v.y;   // x=1: ow=0,rx=1
        A1[kb * 16 + nb + 1]       = v.z;   // x=2: ow=1,rx=0
        A1[(kb + 1) * 16 + nb + 1] = v.w;   // x=3: ow=1,rx=1
    }
    __syncthreads();

    const int wave = t >> 5;
    const int lane = t & 31;
    const int l16  = lane & 15;
    const int hi   = lane >> 4;                      // 0: K{0,1}, 1: K{2,3}
    const float inv = rsqrtf(1.0f + BN_EPS);

    // ---- GEMM1: wave owns M rows [wave*16, wave*16+16) -------------------
    v8f acc = {};
    {
        const int row = wave * 16 + l16;
        const float* wrow = ew1 + (size_t)row * C4;
        for (int ks = 0; ks < C4 / 4; ks++) {
            int k0 = ks * 4 + hi * 2;
            v2f a = *(const v2f*)(wrow + k0);                  // A: W1[row][k0..k0+1]
            v2f bf; bf.x = A1[k0 * 16 + l16];                  // B: act[k0..k0+1][n]
                    bf.y = A1[(k0 + 1) * 16 + l16];
            acc = __builtin_amdgcn_wmma_f32_16x16x4_f32(
                false, a, false, bf, (short)0, acc, false, false);
        }
    }
    // BN + SiLU, write Hs[m][n]  (C/D layout: vgpr i -> rows i and i+8)
    #pragma unroll
    for (int i = 0; i < 8; i++) {
        int m = wave * 16 + i + hi * 8;
        float v = acc[i] * (eg1[m] * inv) + eb1[m];
        Hs[m * 16 + l16] = silu_f(v);
    }
    __syncthreads();

    // ---- depthwise 3x3 (pad 1) on 2x2 grid + BN + SiLU -------------------
    #pragma unroll
    for (int j = 0; j < 8; j++) {
        int e = t + j * 256;                         // 0..2047
        int m = e >> 4;
        int n = e & 15;
        int p = n >> 2, pix = n & 3;
        int oh = pix >> 1, ow = pix & 1;
        const float* wm = wdw + m * 9;
        const float* hb = Hs + m * 16 + p * 4;
        float s = 0.0f;
        #pragma unroll
        for (int iy = 0; iy < 2; iy++)
            #pragma unroll
            for (int ix = 0; ix < 2; ix++)
                s = fmaf(wm[(iy - oh + 1) * 3 + (ix - ow + 1)], hb[iy * 2 + ix], s);
        float v = s * (dwg[m] * inv) + dwb[m];
        A2[m * 16 + n] = silu_f(v);
    }
    __syncthreads();

    // ---- GEMM2 (8 M-tiles per wave) + BN + scale + pixel_shuffle scatter --
    for (int tt = 0; tt < 8; tt++) {
        const int mbase = (wave * 8 + tt) * 16;
        v8f acc2 = {};
        const int row = mbase + l16;
        const float* wrow = ew2 + (size_t)row * HID;
        for (int ks = 0; ks < HID / 4; ks++) {
            int k0 = ks * 4 + hi * 2;
            v2f a = *(const v2f*)(wrow + k0);
            v2f bf; bf.x = A2[k0 * 16 + l16];
                    bf.y = A2[(k0 + 1) * 16 + l16];
            acc2 = __builtin_amdgcn_wmma_f32_16x16x4_f32(
                false, a, false, bf, (short)0, acc2, false, false);
        }
        #pragma unroll
        for (int i = 0; i < 8; i++) {
            int k4 = mbase + i + hi * 8;
            int n  = l16;
            float v = acc2[i] * (eg2[k4] * inv) + eb2[k4];
            int p = n >> 2, pix = n & 3;
            v *= pval[p];
            int c  = k4 >> 2;
            int ry = (k4 >> 1) & 1, rx = k4 & 1;
            int oh = pix >> 1, ow = pix & 1;
            int yy = py0s[p] + oh * 2 + ry;
            int xx = px0s[p] + ow * 2 + rx;
            out[(((size_t)(b * C1 + c)) * IMH + yy) * IMW + xx] = v;
        }
    }
}

// ---------------------------------------------------------------------------
extern "C" void kernel_launch(void* const* d_in, const int* in_sizes, int n_in,
                              void* d_out, int out_size, void* d_ws, size_t ws_size,
                              hipStream_t stream) {
    const float* x      = (const float*)d_in[0];
    const float* rw1    = (const float*)d_in[1];
    const float* rg1    = (const float*)d_in[2];
    const float* rb1    = (const float*)d_in[3];
    const float* rw2    = (const float*)d_in[4];
    const float* rbias2 = (const float*)d_in[5];
    const float* ew1    = (const float*)d_in[6];
    const float* eg1    = (const float*)d_in[7];
    const float* eb1    = (const float*)d_in[8];
    const float* dww    = (const float*)d_in[9];
    const float* dwg    = (const float*)d_in[10];
    const float* dwb    = (const float*)d_in[11];
    const float* ew2    = (const float*)d_in[12];
    const float* eg2    = (const float*)d_in[13];
    const float* eb2    = (const float*)d_in[14];
    float* out = (float*)d_out;

    // workspace layout
    float* pooled  = (float*)d_ws;                               // B*NPATCH*C1
    float* scores  = pooled + (size_t)BATCH * NPATCH * C1;       // B*NPATCH
    int*   sel_idx = (int*)(scores + (size_t)BATCH * NPATCH);    // B*KSEL
    float* sel_val = (float*)(sel_idx + (size_t)BATCH * KSEL);   // B*KSEL

    hipMemsetAsync(d_out, 0, (size_t)out_size * sizeof(float), stream);

    pool_kernel<<<(BATCH * C1 * GH * GW) / 256, 256, 0, stream>>>(x, pooled);
    router_kernel<<<(BATCH * NPATCH) / 8, 256, 0, stream>>>(
        pooled, rw1, rg1, rb1, rw2, rbias2, scores);
    topk_kernel<<<BATCH, 256, 0, stream>>>(scores, sel_idx, sel_val);
    expert_kernel<<<(BATCH * KSEL) / 4, 256, 0, stream>>>(
        x, sel_idx, sel_val, ew1, eg1, eb1, dww, dwg, dwb, ew2, eg2, eb2, out);
}